// DifferentiableFeatureAligner_39152921870905
// MI455X (gfx1250) — compile-verified
//
#include <hip/hip_runtime.h>
#include <math.h>

typedef float v2f __attribute__((ext_vector_type(2)));
typedef float v4f __attribute__((ext_vector_type(4)));
typedef float v8f __attribute__((ext_vector_type(8)));

#define N_POS 16384
#define EMB 128
#define TEMP_INV 10.0f
#define LN_EPSF 1e-5f
#define KSTR 132      // K-tile LDS row stride (floats)
#define PSTR 34       // P / V-tile LDS row stride (floats)

// ---------------- Stage 1: descriptor + MLP encode (Q for ego, K for other) ---------
__global__ __launch_bounds__(64) void encode_kernel(
    const float* __restrict__ ego_cls, const float* __restrict__ ego_reg,
    const float* __restrict__ other_cls, const float* __restrict__ other_reg,
    const float* __restrict__ w1, const float* __restrict__ b1,
    const float* __restrict__ ln_g, const float* __restrict__ ln_b,
    const float* __restrict__ w2, const float* __restrict__ b2,
    float* __restrict__ Qbuf, float* __restrict__ Kbuf)
{
    __shared__ float hls[64 * 129];            // padded stride -> conflict-free
    const int tid = threadIdx.x;
    const int p   = blockIdx.x * 64 + tid;     // token index

    const float* cls; const float* reg; float* outp;
    if (blockIdx.y == 0) { cls = ego_cls;   reg = ego_reg;   outp = Qbuf; }
    else                 { cls = other_cls; reg = other_reg; outp = Kbuf; }

    // descriptor: [size0,size1,size2, sin(th), cos(th), max(cls0,cls1)]
    float x0 = reg[3 * N_POS + p];
    float x1 = reg[4 * N_POS + p];
    float x2 = reg[5 * N_POS + p];
    float th = reg[6 * N_POS + p];
    float x3 = sinf(th);
    float x4 = cosf(th);
    float x5 = fmaxf(cls[p], cls[N_POS + p]);

    float* hrow = &hls[tid * 129];
    float sum = 0.f, sumsq = 0.f;
    for (int c = 0; c < EMB; ++c) {
        float acc = b1[c];
        acc += x0 * w1[0 * EMB + c];
        acc += x1 * w1[1 * EMB + c];
        acc += x2 * w1[2 * EMB + c];
        acc += x3 * w1[3 * EMB + c];
        acc += x4 * w1[4 * EMB + c];
        acc += x5 * w1[5 * EMB + c];
        acc = fmaxf(acc, 0.f);                 // relu
        hrow[c] = acc;
        sum += acc; sumsq += acc * acc;
    }
    float mu   = sum * (1.f / EMB);
    float var  = sumsq * (1.f / EMB) - mu * mu;   // population var (jnp.var)
    float rstd = rsqrtf(var + LN_EPSF);
    for (int c = 0; c < EMB; ++c)
        hrow[c] = (hrow[c] - mu) * rstd * ln_g[c] + ln_b[c];

    for (int e = 0; e < EMB; ++e) {
        float acc = b2[e];
        for (int c = 0; c < EMB; ++c)
            acc += hrow[c] * w2[c * EMB + e];      // w2 addr is wave-uniform
        outp[p * EMB + e] = acc;
    }
}

// ---------------- Stage 2: pack V = [other_cls(2) ; other_reg(14)] as (N,16) --------
__global__ void vpack_kernel(const float* __restrict__ oc, const float* __restrict__ org,
                             float* __restrict__ V)
{
    int t = blockIdx.x * 256 + threadIdx.x;
    if (t >= N_POS * 16) return;
    int p = t >> 4, c = t & 15;
    V[t] = (c < 2) ? oc[c * N_POS + p] : org[(c - 2) * N_POS + p];
}

// ---------------- Stage 3: flash attention, fp32 WMMA, 32-key inner tiles -----------
__global__ __launch_bounds__(256) void attn_kernel(
    const float* __restrict__ Q, const float* __restrict__ K,
    const float* __restrict__ V, float* __restrict__ out)
{
    __shared__ float Kls[32 * KSTR];       // key tile [n][e], 32 keys
    __shared__ float Vls[16 * PSTR];       // V tile transposed [chan][k], k<32
    __shared__ float Pls[8][16 * PSTR];    // per-wave P (16 q x 32 k) transpose scratch

    const int tid  = threadIdx.x;
    const int lane = tid & 31;
    const int wave = tid >> 5;
    const int half = lane >> 4;            // 0: lanes 0-15, 1: lanes 16-31
    const int l16  = lane & 15;
    const int qRow = blockIdx.x * 128 + wave * 16 + l16;

    // Load Q A-fragments once; fold 1/TEMP into Q.
    // A layout (16x4 f32): M = l16, K = kk*4 + 2*half + j  (j = vgpr in pair)
    v2f qf[32];
#pragma unroll
    for (int kk = 0; kk < 32; ++kk) {
        const float* s = &Q[qRow * EMB + kk * 4 + half * 2];
        v2f t; t.x = s[0] * TEMP_INV; t.y = s[1] * TEMP_INV;
        qf[kk] = t;
    }

    v8f Oacc = {};
    float mstate[8], lstate[8];
#pragma unroll
    for (int v = 0; v < 8; ++v) { mstate[v] = -3.0e38f; lstate[v] = 0.f; }

    float* pmine = &Pls[wave][0];

    for (int kt = 0; kt < N_POS / 32; ++kt) {
        __syncthreads();
        // cooperative K-tile load: 32 rows x 128 feats = 1024 float4
#pragma unroll
        for (int i = 0; i < 4; ++i) {
            int idx4 = tid * 4 + i;            // 0..1023
            int row  = idx4 >> 5;              // 32 float4 per row
            int col4 = (idx4 & 31) * 4;
            *(v4f*)&Kls[row * KSTR + col4] = *(const v4f*)&K[(kt * 32 + row) * EMB + col4];
        }
        // cooperative V-tile (transposed) load: 32 keys x 16 chans
#pragma unroll
        for (int i = 0; i < 2; ++i) {
            int idx = tid + i * 256;           // 0..511
            int k = idx >> 4, c = idx & 15;
            Vls[c * PSTR + k] = V[(kt * 32 + k) * 16 + c];
        }
        __syncthreads();

        // ---- S0 = Q' @ K[0:16]^T ; S1 = Q' @ K[16:32]^T (batched frag loads) ----
        v8f S0 = {}, S1 = {};
        {
            v2f bf[32];
#pragma unroll
            for (int kk = 0; kk < 32; ++kk)
                bf[kk] = *(const v2f*)&Kls[l16 * KSTR + kk * 4 + half * 2];
#pragma unroll
            for (int kk = 0; kk < 32; ++kk)
                S0 = __builtin_amdgcn_wmma_f32_16x16x4_f32(false, qf[kk], false, bf[kk],
                                                           (short)0, S0, false, false);
#pragma unroll
            for (int kk = 0; kk < 32; ++kk)
                bf[kk] = *(const v2f*)&Kls[(16 + l16) * KSTR + kk * 4 + half * 2];
#pragma unroll
            for (int kk = 0; kk < 32; ++kk)
                S1 = __builtin_amdgcn_wmma_f32_16x16x4_f32(false, qf[kk], false, bf[kk],
                                                           (short)0, S1, false, false);
        }

        // ---- online softmax over 32 keys, D layout: row m = v + 8*half ----
#pragma unroll
        for (int v = 0; v < 8; ++v) {
            float s0 = S0[v], s1 = S1[v];
            float rm = fmaxf(s0, s1);
            rm = fmaxf(rm, __shfl_xor(rm, 1, 16));
            rm = fmaxf(rm, __shfl_xor(rm, 2, 16));
            rm = fmaxf(rm, __shfl_xor(rm, 4, 16));
            rm = fmaxf(rm, __shfl_xor(rm, 8, 16));
            float newm  = fmaxf(mstate[v], rm);
            float alpha = __expf(mstate[v] - newm);
            float p0    = __expf(s0 - newm);
            float p1    = __expf(s1 - newm);
            float rs = p0 + p1;
            rs += __shfl_xor(rs, 1, 16);
            rs += __shfl_xor(rs, 2, 16);
            rs += __shfl_xor(rs, 4, 16);
            rs += __shfl_xor(rs, 8, 16);
            lstate[v] = lstate[v] * alpha + rs;
            mstate[v] = newm;
            Oacc[v]  *= alpha;
            int m = v + 8 * half;
            pmine[m * PSTR + l16]      = p0;   // keys 0..15
            pmine[m * PSTR + 16 + l16] = p1;   // keys 16..31
        }
        // intra-wave LDS RAW fence for the P transpose
        asm volatile("s_wait_dscnt 0" ::: "memory");

        // ---- O += P @ V : K = 32 keys in 8 chunks of 4 (batched frag loads) ----
        {
            v2f pa[8], vb[8];
#pragma unroll
            for (int c4 = 0; c4 < 8; ++c4) {
                pa[c4] = *(const v2f*)&pmine[l16 * PSTR + c4 * 4 + half * 2];
                vb[c4] = *(const v2f*)&Vls  [l16 * PSTR + c4 * 4 + half * 2];
            }
#pragma unroll
            for (int c4 = 0; c4 < 8; ++c4)
                Oacc = __builtin_amdgcn_wmma_f32_16x16x4_f32(false, pa[c4], false, vb[c4],
                                                             (short)0, Oacc, false, false);
        }
    }

    // finalize & write: out[chan*N + q], chan = l16, q row = v + 8*half
#pragma unroll
    for (int v = 0; v < 8; ++v) {
        int q = blockIdx.x * 128 + wave * 16 + v + 8 * half;
        out[l16 * N_POS + q] = Oacc[v] / lstate[v];
    }
}

extern "C" void kernel_launch(void* const* d_in, const int* in_sizes, int n_in,
                              void* d_out, int out_size, void* d_ws, size_t ws_size,
                              hipStream_t stream)
{
    const float* ego_cls   = (const float*)d_in[0];
    const float* ego_reg   = (const float*)d_in[1];
    const float* other_cls = (const float*)d_in[2];
    const float* other_reg = (const float*)d_in[3];
    const float* w1  = (const float*)d_in[4];
    const float* b1  = (const float*)d_in[5];
    const float* ln_g = (const float*)d_in[6];
    const float* ln_b = (const float*)d_in[7];
    const float* w2  = (const float*)d_in[8];
    const float* b2  = (const float*)d_in[9];

    float* Qbuf = (float*)d_ws;                 // 16384*128 f32
    float* Kbuf = Qbuf + N_POS * EMB;           // 16384*128 f32
    float* Vbuf = Kbuf + N_POS * EMB;           // 16384*16  f32
    float* out  = (float*)d_out;

    encode_kernel<<<dim3(N_POS / 64, 2), 64, 0, stream>>>(
        ego_cls, ego_reg, other_cls, other_reg,
        w1, b1, ln_g, ln_b, w2, b2, Qbuf, Kbuf);

    vpack_kernel<<<(N_POS * 16 + 255) / 256, 256, 0, stream>>>(other_cls, other_reg, Vbuf);

    attn_kernel<<<N_POS / 128, 256, 0, stream>>>(Qbuf, Kbuf, Vbuf, out);
}